// CrossSourceDomainMP_70858370450102
// MI455X (gfx1250) — compile-verified
//
#include <hip/hip_runtime.h>
#include <hip/hip_bf16.h>
#include <math.h>

// ---------------------------------------------------------------------------
// CrossSourceDomainMP for MI455X (gfx1250, wave32).
// Hot path: fused cross-domain cosine-sim GEMM + row top-5, using
// V_WMMA_F32_16X16X4_F32 (exact f32 matrix path -> faithful kNN selection),
// with 4-way column-tile blocking per wave (shared A fragments, four
// independent wmma accumulator chains) and one-step software pipelining of
// the B fragments so ~8 global_load_b64 stay in flight ahead of the wmmas.
// ---------------------------------------------------------------------------

#define N_DOM0 6000
#define N_DOM1 6000
#define NTOT   12000
#define H      256
#define KNN    5
#define NEDGE  (NTOT * KNN)

typedef __attribute__((ext_vector_type(2))) float v2f;
typedef __attribute__((ext_vector_type(8))) float v8f;

// ---------------------------------------------------------------------------
// Kernel 1: concatenate embeddings and row-normalize (v = emb / max(||emb||,eps))
// One wave per row; 8 rows per 256-thread block.
// ---------------------------------------------------------------------------
__global__ __launch_bounds__(256) void k_normalize(
    const float* __restrict__ e0, const float* __restrict__ e1,
    float* __restrict__ embC, float* __restrict__ v)
{
  const int row  = blockIdx.x * 8 + (threadIdx.x >> 5);
  const int lane = threadIdx.x & 31;
  if (row >= NTOT) return;
  const float* src = (row < N_DOM0) ? (e0 + (size_t)row * H)
                                    : (e1 + (size_t)(row - N_DOM0) * H);
  float vals[8];
  float ss = 0.0f;
#pragma unroll
  for (int i = 0; i < 8; ++i) {
    float x = src[lane + i * 32];
    vals[i] = x;
    ss += x * x;
  }
#pragma unroll
  for (int off = 16; off >= 1; off >>= 1) ss += __shfl_xor(ss, off, 32);
  float nrm = fmaxf(sqrtf(ss), 1e-12f);
  float inv = 1.0f / nrm;
  float* ec = embC + (size_t)row * H;
  float* vr = v    + (size_t)row * H;
#pragma unroll
  for (int i = 0; i < 8; ++i) {
    ec[lane + i * 32] = vals[i];
    vr[lane + i * 32] = vals[i] * inv;
  }
}

// ---------------------------------------------------------------------------
// Kernel 2: fused similarity GEMM + top-5 per row (cross-domain block only;
// same-domain block is masked to zero in the reference so we skip it).
// One block per 16-row stripe (750 blocks), 8 waves; each wave owns groups of
// four 16x16 column tiles (16x64 C block, four wmma chains sharing A frags),
// with B fragments prefetched one k-step ahead. K=256 -> 4x64 wmma per group.
// ---------------------------------------------------------------------------
__global__ __launch_bounds__(256) void k_sim_topk(
    const float* __restrict__ v,
    float* __restrict__ topVal, int* __restrict__ topIdx)
{
  __shared__ float As[16][H];            // 16 KB : A stripe (row-major)
  __shared__ float tileS[8][16][65];     // per-wave 16x64 C block (padded)
  __shared__ float mVal[8][32][KNN];     // per-lane top-5 merge buffer
  __shared__ int   mIdx[8][32][KNN];

  const int tid   = threadIdx.x;
  const int wave  = tid >> 5;
  const int lane  = tid & 31;
  const int row0  = blockIdx.x * 16;

  // stage A stripe: 16x256 floats, coalesced
  for (int i = tid; i < 16 * H; i += 256)
    As[i >> 8][i & 255] = v[(size_t)(row0 + (i >> 8)) * H + (i & 255)];
  __syncthreads();

  const bool dom0    = (row0 < N_DOM0);
  const int colBase  = dom0 ? N_DOM0 : 0;
  const int numTiles = (dom0 ? N_DOM1 : N_DOM0) >> 4;   // 375

  // per-lane running top-5 (sorted descending)
  float tv[KNN]; int ti[KNN];
#pragma unroll
  for (int k = 0; k < KNN; ++k) { tv[k] = -1e30f; ti[k] = 0; }

  const int m     = lane & 15;          // A row / B row this lane loads
  const int koff  = (lane >> 4) * 2;    // K sub-offset per 32-bit frag layout
  const int rrow  = lane >> 1;          // row scanned in top-k phase
  const int chalf = (lane & 1) * 8;     // column half scanned

  for (int t = wave * 4; t < numTiles; t += 32) {
    const float* __restrict__ Bp[4];
#pragma unroll
    for (int q = 0; q < 4; ++q) {
      const int tq = (t + q < numTiles) ? (t + q) : t;   // clamp tail
      Bp[q] = v + (size_t)(colBase + tq * 16 + m) * H;
    }
    // prefetch next tile group's B rows for this lane (global_prefetch_b8)
    if (t + 32 < numTiles)
      __builtin_prefetch(v + (size_t)(colBase + (t + 32) * 16 + m) * H, 0, 1);

    v8f c[4] = {{}, {}, {}, {}};

    // software-pipelined fragment stream: load k-step ahead of compute
    v2f a_c, b_c[4];
    a_c.x = As[m][koff];
    a_c.y = As[m][koff + 1];
#pragma unroll
    for (int q = 0; q < 4; ++q) b_c[q] = *(const v2f*)(Bp[q] + koff);

#pragma unroll
    for (int kb = 0; kb < H; kb += 4) {
      v2f a_n = {}, b_n[4] = {{}, {}, {}, {}};
      if (kb + 4 < H) {                       // compile-time (full unroll)
        a_n.x = As[m][kb + 4 + koff];
        a_n.y = As[m][kb + 4 + koff + 1];
#pragma unroll
        for (int q = 0; q < 4; ++q)
          b_n[q] = *(const v2f*)(Bp[q] + kb + 4 + koff);
      }
#pragma unroll
      for (int q = 0; q < 4; ++q)
        c[q] = __builtin_amdgcn_wmma_f32_16x16x4_f32(false, a_c, false, b_c[q],
                                                     (short)0, c[q], false, false);
      a_c = a_n;
#pragma unroll
      for (int q = 0; q < 4; ++q) b_c[q] = b_n[q];
    }

    // spill C block through LDS (lane = col n, VGPR j = row m0+j)
    const int n  = lane & 15;
    const int m0 = (lane >> 4) * 8;
#pragma unroll
    for (int q = 0; q < 4; ++q)
#pragma unroll
      for (int j = 0; j < 8; ++j)
        tileS[wave][m0 + j][q * 16 + n] = c[q][j];
    __builtin_amdgcn_wave_barrier();   // DS ops are in-order per wave

    // top-5 update: 2 lanes per row, 8 columns per sub-tile each
#pragma unroll
    for (int q = 0; q < 4; ++q) {
      if (t + q < numTiles) {
#pragma unroll
        for (int cc = 0; cc < 8; ++cc) {
          float val = tileS[wave][rrow][q * 16 + chalf + cc];
          if (val > tv[KNN - 1]) {
            int idx = colBase + (t + q) * 16 + chalf + cc;
            int p = KNN - 1;
            while (p > 0 && val > tv[p - 1]) {
              tv[p] = tv[p - 1]; ti[p] = ti[p - 1]; --p;
            }
            tv[p] = val; ti[p] = idx;
          }
        }
      }
    }
    __builtin_amdgcn_wave_barrier();
  }

#pragma unroll
  for (int k = 0; k < KNN; ++k) { mVal[wave][lane][k] = tv[k]; mIdx[wave][lane][k] = ti[k]; }
  __syncthreads();

  // final merge: thread r (<16) merges 8 waves x 2 lanes x 5 candidates
  if (tid < 16) {
    float bv[KNN]; int bi[KNN];
#pragma unroll
    for (int k = 0; k < KNN; ++k) { bv[k] = -1e30f; bi[k] = 0; }
    for (int w = 0; w < 8; ++w)
      for (int l2 = 0; l2 < 2; ++l2)
#pragma unroll
        for (int k = 0; k < KNN; ++k) {
          float val = mVal[w][tid * 2 + l2][k];
          if (val > bv[KNN - 1]) {
            int idx = mIdx[w][tid * 2 + l2][k];
            int p = KNN - 1;
            while (p > 0 && val > bv[p - 1]) {
              bv[p] = bv[p - 1]; bi[p] = bi[p - 1]; --p;
            }
            bv[p] = val; bi[p] = idx;
          }
        }
    const int gr = row0 + tid;
#pragma unroll
    for (int k = 0; k < KNN; ++k) {
      topVal[gr * KNN + k] = bv[k];
      topIdx[gr * KNN + k] = bi[k];
    }
  }
}

// ---------------------------------------------------------------------------
// Kernel 3: xw = emb @ gnn_weight^T  (12000x256 @ 256x256), same WMMA scheme.
// One block per 16-row stripe; each of 8 waves owns 2 column tiles with
// shared A fragments and two wmma chains.
// ---------------------------------------------------------------------------
__global__ __launch_bounds__(256) void k_gemm_xw(
    const float* __restrict__ embC, const float* __restrict__ gw,
    float* __restrict__ xw)
{
  __shared__ float As[16][H];
  const int tid  = threadIdx.x;
  const int wave = tid >> 5;
  const int lane = tid & 31;
  const int row0 = blockIdx.x * 16;

  for (int i = tid; i < 16 * H; i += 256)
    As[i >> 8][i & 255] = embC[(size_t)(row0 + (i >> 8)) * H + (i & 255)];
  __syncthreads();

  const int m    = lane & 15;
  const int koff = (lane >> 4) * 2;
  const int t0   = wave * 2;            // two of 16 column tiles per wave
  const int t1   = t0 + 1;

  // B[k][n] = gnn_weight[t*16+n][k]  (xw = x @ W^T)
  const float* __restrict__ Bp0 = gw + (size_t)(t0 * 16 + m) * H;
  const float* __restrict__ Bp1 = gw + (size_t)(t1 * 16 + m) * H;

  v8f c0 = {}, c1 = {};
#pragma unroll
  for (int kb = 0; kb < H; kb += 4) {
    v2f a;
    a.x = As[m][kb + koff];
    a.y = As[m][kb + koff + 1];
    const v2f b0 = *(const v2f*)(Bp0 + kb + koff);
    const v2f b1 = *(const v2f*)(Bp1 + kb + koff);
    c0 = __builtin_amdgcn_wmma_f32_16x16x4_f32(false, a, false, b0,
                                               (short)0, c0, false, false);
    c1 = __builtin_amdgcn_wmma_f32_16x16x4_f32(false, a, false, b1,
                                               (short)0, c1, false, false);
  }
  const int n  = lane & 15;
  const int m0 = (lane >> 4) * 8;
#pragma unroll
  for (int j = 0; j < 8; ++j) {
    xw[(size_t)(row0 + m0 + j) * H + t0 * 16 + n] = c0[j];
    xw[(size_t)(row0 + m0 + j) * H + t1 * 16 + n] = c1[j];
  }
}

// ---------------------------------------------------------------------------
// Kernel 4: init scratch (gcnOut=0, deg=1 self-loop, acc=0) — every call.
// ---------------------------------------------------------------------------
__global__ __launch_bounds__(256) void k_init(
    float* __restrict__ gcnOut, float* __restrict__ deg, float* __restrict__ acc)
{
  const int idx = blockIdx.x * 256 + threadIdx.x;
  if (idx < NTOT * H) gcnOut[idx] = 0.0f;
  if (idx < NTOT)     deg[idx]    = 1.0f;
  if (idx < 2 * H)    acc[idx]    = 0.0f;
}

// Kernel 5: deg[t] += w  over candidate edges
__global__ __launch_bounds__(256) void k_deg(
    const float* __restrict__ topVal, const int* __restrict__ topIdx,
    float* __restrict__ deg)
{
  const int e = blockIdx.x * 256 + threadIdx.x;
  if (e >= NEDGE) return;
  float w = (topVal[e] != 0.0f) ? 1.0f : 0.0f;
  if (w != 0.0f) atomicAdd(&deg[topIdx[e]], w);
}

// Kernel 6: deg -> dinv (in place)
__global__ __launch_bounds__(256) void k_dinv(float* __restrict__ deg)
{
  const int i = blockIdx.x * 256 + threadIdx.x;
  if (i >= NTOT) return;
  float d = deg[i];
  deg[i] = (d > 0.0f) ? rsqrtf(fmaxf(d, 1e-30f)) : 0.0f;
}

// Kernel 7: message scatter — one wave per edge, 8 f32 per lane, f32 atomics.
__global__ __launch_bounds__(256) void k_msg(
    const float* __restrict__ topVal, const int* __restrict__ topIdx,
    const float* __restrict__ dinv, const float* __restrict__ xw,
    float* __restrict__ gcnOut)
{
  const int e    = blockIdx.x * 8 + (threadIdx.x >> 5);
  const int lane = threadIdx.x & 31;
  if (e >= NEDGE) return;
  if (topVal[e] == 0.0f) return;           // ew == 0 contributes nothing
  const int s = e / KNN;
  const int t = topIdx[e];
  const float nrm = dinv[s] * dinv[t];
  const float* __restrict__ xs = xw + (size_t)s * H;
  float* __restrict__ ot = gcnOut + (size_t)t * H;
#pragma unroll
  for (int i = 0; i < 8; ++i)
    atomicAdd(&ot[lane + i * 32], nrm * xs[lane + i * 32]);
}

// ---------------------------------------------------------------------------
// Kernel 8: self-loop + bias + 2-way attention softmax; writes updated to
// d_out[1:]. One wave per row.
// ---------------------------------------------------------------------------
__global__ __launch_bounds__(256) void k_attn(
    const float* __restrict__ embC, const float* __restrict__ gcnOut,
    const float* __restrict__ xw, const float* __restrict__ dinv,
    const float* __restrict__ gb, const float* __restrict__ aw,
    const float* __restrict__ ab, float* __restrict__ outUpd)
{
  const int row  = blockIdx.x * 8 + (threadIdx.x >> 5);
  const int lane = threadIdx.x & 31;
  if (row >= NTOT) return;
  const float self = dinv[row] * dinv[row];
  const float* er = embC   + (size_t)row * H;
  const float* gr = gcnOut + (size_t)row * H;
  const float* xr = xw     + (size_t)row * H;

  float ev[8], gv[8];
  float s0 = 0.0f, s1 = 0.0f;
#pragma unroll
  for (int i = 0; i < 8; ++i) {
    const int c = lane + i * 32;
    float e = er[c];
    float g = gr[c] + self * xr[c] + gb[c];
    ev[i] = e; gv[i] = g;
    float w = aw[c];
    s0 += e * w;
    s1 += g * w;
  }
#pragma unroll
  for (int off = 16; off >= 1; off >>= 1) {
    s0 += __shfl_xor(s0, off, 32);
    s1 += __shfl_xor(s1, off, 32);
  }
  const float b = ab[0];
  s0 += b; s1 += b;
  const float mx = fmaxf(s0, s1);
  const float w0 = __expf(s0 - mx);
  const float w1 = __expf(s1 - mx);
  const float inv = 1.0f / (w0 + w1);
  float* orow = outUpd + (size_t)row * H;
#pragma unroll
  for (int i = 0; i < 8; ++i) {
    const int c = lane + i * 32;
    orow[c] = (w0 * ev[i] + w1 * gv[i]) * inv;
  }
}

// Kernel 9: per-domain column sums of updated (for MMD)
__global__ __launch_bounds__(256) void k_colsum(
    const float* __restrict__ upd, float* __restrict__ acc)
{
  const int c  = threadIdx.x;         // column 0..255
  const int r0 = blockIdx.x * 256;
  float s0 = 0.0f, s1 = 0.0f;
  for (int i = 0; i < 256; ++i) {
    const int r = r0 + i;
    if (r >= NTOT) break;
    float v = upd[(size_t)r * H + c];
    if (r < N_DOM0) s0 += v; else s1 += v;
  }
  if (s0 != 0.0f) atomicAdd(&acc[c], s0);
  if (s1 != 0.0f) atomicAdd(&acc[H + c], s1);
}

// Kernel 10: loss = sum_c (mean0[c] - mean1[c])^2  -> d_out[0]
__global__ __launch_bounds__(256) void k_loss(
    const float* __restrict__ acc, float* __restrict__ outLoss)
{
  __shared__ float part[8];
  const int c    = threadIdx.x;
  const int wave = c >> 5;
  const int lane = c & 31;
  float d = acc[c] * (1.0f / N_DOM0) - acc[H + c] * (1.0f / N_DOM1);
  float p = d * d;
#pragma unroll
  for (int off = 16; off >= 1; off >>= 1) p += __shfl_xor(p, off, 32);
  if (lane == 0) part[wave] = p;
  __syncthreads();
  if (c == 0) {
    float s = 0.0f;
#pragma unroll
    for (int w = 0; w < 8; ++w) s += part[w];
    outLoss[0] = s;
  }
}

// ---------------------------------------------------------------------------
extern "C" void kernel_launch(void* const* d_in, const int* in_sizes, int n_in,
                              void* d_out, int out_size, void* d_ws, size_t ws_size,
                              hipStream_t stream)
{
  (void)in_sizes; (void)n_in; (void)out_size; (void)ws_size;

  const float* e0 = (const float*)d_in[2];   // src_embedding_0 (6000x256)
  const float* e1 = (const float*)d_in[3];   // src_embedding_1 (6000x256)
  const float* gw = (const float*)d_in[8];   // gnn_weight (256x256)
  const float* gb = (const float*)d_in[9];   // gnn_bias (256)
  const float* aw = (const float*)d_in[10];  // attn_weight (1x256)
  const float* ab = (const float*)d_in[11];  // attn_bias (1)

  float* out    = (float*)d_out;             // [0]=loss, [1:]=updated
  float* ws     = (float*)d_ws;
  float* v      = ws;
  float* embC   = v      + (size_t)NTOT * H;
  float* xw     = embC   + (size_t)NTOT * H;
  float* gcnOut = xw     + (size_t)NTOT * H;
  float* topVal = gcnOut + (size_t)NTOT * H;
  int*   topIdx = (int*)(topVal + NEDGE);
  float* deg    = (float*)(topIdx + NEDGE);  // becomes dinv in-place
  float* acc    = deg + NTOT;

  float* outUpd = out + 1;

  k_normalize<<<NTOT / 8, 256, 0, stream>>>(e0, e1, embC, v);
  k_sim_topk <<<NTOT / 16, 256, 0, stream>>>(v, topVal, topIdx);
  k_gemm_xw  <<<NTOT / 16, 256, 0, stream>>>(embC, gw, xw);
  k_init     <<<(NTOT * H + 255) / 256, 256, 0, stream>>>(gcnOut, deg, acc);
  k_deg      <<<(NEDGE + 255) / 256, 256, 0, stream>>>(topVal, topIdx, deg);
  k_dinv     <<<(NTOT + 255) / 256, 256, 0, stream>>>(deg);
  k_msg      <<<NEDGE / 8, 256, 0, stream>>>(topVal, topIdx, deg, xw, gcnOut);
  k_attn     <<<NTOT / 8, 256, 0, stream>>>(embC, gcnOut, xw, deg, gb, aw, ab, outUpd);
  k_colsum   <<<(NTOT + 255) / 256, 256, 0, stream>>>(outUpd, acc);
  k_loss     <<<1, 256, 0, stream>>>(acc, out);
}